// Node_3977139716155
// MI455X (gfx1250) — compile-verified
//
#include <hip/hip_runtime.h>
#include <stdint.h>

// ---------------------------------------------------------------------------
// out[n] = sigmoid( concat(values[nbr[n,:]]) @ W[n] + b[n] )
//   N=200000, K=16, D=8  ->  per-node GEMV [1x128]@[128x8]
// Memory-bound: ~940 MB @ 23.3 TB/s => ~40us floor; 0.48 FLOP/byte.
//
// Design (wave32, 2 nodes per wave-iteration):
//  * neighbor gather: one global_load_async_to_lds_b128 per node (per-lane
//    scattered global addresses -> linear LDS), double buffered (ASYNCcnt).
//  * W[n] streamed directly with NT b32 loads (sequential 4KB per node).
//  * lane l computes 2 partial dots (32 features each) for output slot
//    m = l&15 (= node_in_pair*8 + d); K-half = l>>4.
//  * ONE v_wmma_f32_16x16x4_f32 with B=ones sums the 4 K-slots per row:
//    D[m,*] = full 128-feature dot product, exact f32. The matrix unit does
//    the whole cross-wave reduction (replaces a 5-round shfl_xor butterfly).
//  * lanes 0/16 add bias, sigmoid (v_exp + v_rcp), store 32B each.
// ---------------------------------------------------------------------------

#define N_NODES   200000
#define N_PAIRS   (N_NODES / 2)
#define K_NBR     16
#define D_DIM     8
#define W_FLOATS  1024            // 128*8 floats = 4KB per node
#define WAVE      32
#define WPB       8               // waves per block
#define TPB       (WPB * WAVE)
#define NBLOCKS   2500            // 20000 waves -> 5 pairs per wave

// Padded gathered-values layout in LDS (per node): two 64-float halves,
// each half padded 256B->272B; node stride 544B. Decorrelates the 4
// broadcast address groups (nd,h) onto disjoint LDS bank groups.
#define G_HALF_B  272
#define G_NODE_B  544
#define G_PAIR_B  (2 * G_NODE_B)  // 1088B per pair buffer

typedef __attribute__((ext_vector_type(2))) float v2f;
typedef __attribute__((ext_vector_type(8))) float v8f;

// CDNA5 async DMA: 16B per lane, global -> LDS, tracked by ASYNCcnt.
__device__ __forceinline__ void async_ld_b128(uint32_t lds_byte, unsigned long long gaddr) {
  asm volatile("global_load_async_to_lds_b128 %0, %1, off th:TH_LOAD_NT"
               :: "v"(lds_byte), "v"(gaddr) : "memory");
}
__device__ __forceinline__ void wait_async0() { asm volatile("s_wait_asynccnt 0x0" ::: "memory"); }
__device__ __forceinline__ void wait_async2() { asm volatile("s_wait_asynccnt 0x2" ::: "memory"); }
__device__ __forceinline__ void wait_ds0()    { asm volatile("s_wait_dscnt 0x0"    ::: "memory"); }

// sigmoid via v_exp_f32 + v_rcp_f32 (avoid the IEEE-divide fixup chain).
__device__ __forceinline__ float fast_sigmoid(float x) {
#if __has_builtin(__builtin_amdgcn_rcpf)
  return __builtin_amdgcn_rcpf(1.0f + __expf(-x));
#else
  return __frcp_rn(1.0f + __expf(-x));
#endif
}

// Async-gather both nodes of a pair into the LDS buffer at ldsBase.
// Lane l fetches g[f=4l..4l+3] = values[nbr[f>>3]][f&7 ..], 16B per lane.
__device__ __forceinline__ void gather_pair(const long long* __restrict__ nidx,
                                            const float* __restrict__ values,
                                            int pairIdx, uint32_t ldsBase, int lane) {
  #pragma unroll
  for (int nd = 0; nd < 2; ++nd) {
    const long long gi = nidx[(size_t)(2 * pairIdx + nd) * K_NBR + (lane >> 1)];
    const unsigned long long ga =
        (unsigned long long)(uintptr_t)(values + (size_t)gi * D_DIM + (size_t)(lane & 1) * 4);
    const uint32_t la =
        ldsBase + (uint32_t)(nd * G_NODE_B + (lane >> 4) * G_HALF_B + (lane & 15) * 16);
    async_ld_b128(la, ga);
  }
}

__global__ void __launch_bounds__(TPB)
node_gather_linear_sigmoid(const float* __restrict__ values,
                           const long long* __restrict__ nidx,   // jnp.int64 per reference
                           const float* __restrict__ W,
                           const float* __restrict__ bias,
                           float* __restrict__ out)
{
  // 8 waves * 2 buffers * 1088B = 17KB LDS per block
  __shared__ char gmem[WPB][2][G_PAIR_B];

  const int lane   = (int)threadIdx.x & (WAVE - 1);
  const int wib    = (int)threadIdx.x / WAVE;
  const int wave   = (int)blockIdx.x * WPB + wib;
  const int nwaves = (int)gridDim.x * WPB;

  const int d  = lane & 7;          // output dim owned by this lane
  const int nd = (lane >> 3) & 1;   // node-in-pair owned by this lane
  const int h  = lane >> 4;         // K-half: f in [64h, 64h+64)

  const uint32_t ldsB0 = (uint32_t)(uintptr_t)&gmem[wib][0][0];
  const uint32_t ldsB1 = (uint32_t)(uintptr_t)&gmem[wib][1][0];

  int pair = wave;
  if (pair < N_PAIRS)
    gather_pair(nidx, values, pair, ldsB0, lane);   // prologue prefetch

  int buf = 0;
  for (; pair < N_PAIRS; pair += nwaves) {
    const int  nextPair = pair + nwaves;
    const bool hasNext  = nextPair < N_PAIRS;

    // Prefetch next pair's gather into the other buffer. Its previous reads
    // retired last iteration; dscnt fence makes that explicit.
    if (hasNext) {
      wait_ds0();
      gather_pair(nidx, values, nextPair, buf ? ldsB0 : ldsB1, lane);
    }

    // W stream: this lane reads W[node][f][d], f = 64h + t, stride 8 floats.
    const float* wbase =
        W + (size_t)(2 * pair + nd) * W_FLOATS + (size_t)(64 * h) * D_DIM + d;

    // Wait for current gather (allow next pair's 2 copies in flight).
    if (hasNext) wait_async2(); else wait_async0();

    const char* gb = (const char*)&gmem[wib][buf][0] + nd * G_NODE_B + h * G_HALF_B;

    float p0 = 0.f, p1 = 0.f;   // partials over f halves [64h,+32) and [64h+32,+32)
    #pragma unroll
    for (int jc = 0; jc < 8; ++jc) {
      const float4 ga4 = *(const float4*)(gb + 16 * jc);        // g[64h+4jc .. +3]
      const float4 gb4 = *(const float4*)(gb + 128 + 16 * jc);  // g[64h+32+4jc .. +3]
      const float gav[4] = { ga4.x, ga4.y, ga4.z, ga4.w };
      const float gbv[4] = { gb4.x, gb4.y, gb4.z, gb4.w };
      #pragma unroll
      for (int t = 0; t < 4; ++t) {
        const float w0 = __builtin_nontemporal_load(wbase + (size_t)(4 * jc + t) * D_DIM);
        const float w1 = __builtin_nontemporal_load(wbase + (size_t)(32 + 4 * jc + t) * D_DIM);
        p0 = __builtin_fmaf(gav[t], w0, p0);
        p1 = __builtin_fmaf(gbv[t], w1, p1);
      }
    }

    // WMMA reduction: A[m=l&15, k] = partials (k = 2h + {0,1}), B = ones.
    // D[m,n] = sum_k A[m,k]  (same for every n)  = full dot product for slot m.
    v2f a; a[0] = p0; a[1] = p1;
    v2f bones; bones[0] = 1.0f; bones[1] = 1.0f;
    v8f c = {};
    const v8f dv = __builtin_amdgcn_wmma_f32_16x16x4_f32(
        /*neg_a=*/false, a, /*neg_b=*/false, bones,
        /*c_mod=*/(short)0, c, /*reuse_a=*/false, /*reuse_b=*/false);

    // lane 0 holds M=0..7 (node 2*pair, d=0..7); lane 16 holds M=8..15 (node 2*pair+1).
    if ((lane & 15) == 0) {
      const int myNode = 2 * pair + (lane >> 4);
      const float4 b0 = *(const float4*)(bias + (size_t)myNode * D_DIM);
      const float4 b1 = *(const float4*)(bias + (size_t)myNode * D_DIM + 4);
      float4 o0, o1;
      o0.x = fast_sigmoid(dv[0] + b0.x);
      o0.y = fast_sigmoid(dv[1] + b0.y);
      o0.z = fast_sigmoid(dv[2] + b0.z);
      o0.w = fast_sigmoid(dv[3] + b0.w);
      o1.x = fast_sigmoid(dv[4] + b1.x);
      o1.y = fast_sigmoid(dv[5] + b1.y);
      o1.z = fast_sigmoid(dv[6] + b1.z);
      o1.w = fast_sigmoid(dv[7] + b1.w);
      *(float4*)(out + (size_t)myNode * D_DIM)     = o0;
      *(float4*)(out + (size_t)myNode * D_DIM + 4) = o1;
    }

    buf ^= 1;
  }
}

extern "C" void kernel_launch(void* const* d_in, const int* in_sizes, int n_in,
                              void* d_out, int out_size, void* d_ws, size_t ws_size,
                              hipStream_t stream) {
  (void)in_sizes; (void)n_in; (void)d_ws; (void)ws_size; (void)out_size;
  const float*     values = (const float*)d_in[0];
  const long long* nidx   = (const long long*)d_in[1];  // jnp.int64 per reference
  const float*     W      = (const float*)d_in[2];
  const float*     bias   = (const float*)d_in[3];
  float*           out    = (float*)d_out;

  node_gather_linear_sigmoid<<<NBLOCKS, TPB, 0, stream>>>(values, nidx, W, bias, out);
}